// ROIPooler_33380485825264
// MI455X (gfx1250) — compile-verified
//
#include <hip/hip_runtime.h>
#include <hip/hip_bf16.h>

// ---------------------------------------------------------------------------
// Stage 1: NCHW -> NHWC transpose (per pyramid level), staged through LDS
// using CDNA5 async global->LDS loads (ASYNCcnt path).
//   src: [N=2, C=256, HW]   dst: [N=2, HW, C=256]
// ---------------------------------------------------------------------------
__global__ __launch_bounds__(256) void nchw_to_nhwc(const float* __restrict__ src,
                                                    float* __restrict__ dst,
                                                    int HW) {
    __shared__ float tile[32][33];   // +1 pad: conflict-free transposed reads

    const int tx  = threadIdx.x;            // 0..31 (lane)
    const int ty  = threadIdx.y;            // 0..7  (wave id)
    const int hw0 = blockIdx.x * 32;
    const int c0  = blockIdx.y * 32;
    const int n   = blockIdx.z;

    const float* s = src + (size_t)n * 256 * HW;
    float*       d = dst + (size_t)n * HW * 256;

    // Load phase: coalesced 128B/wave reads, written straight into LDS by the
    // async copy unit (no VGPR round trip). OOB hw lanes clamp the source
    // address (value never stored in the write phase).
#pragma unroll
    for (int j = 0; j < 4; ++j) {
        const int c  = c0 + ty + j * 8;
        int hw = hw0 + tx;
        if (hw >= HW) hw = HW - 1;
        const float* gp = s + (size_t)c * HW + hw;
        const unsigned lds = (unsigned)(unsigned long long)(uintptr_t)(&tile[ty + j * 8][tx]);
        asm volatile("global_load_async_to_lds_b32 %0, %1, off"
                     :: "v"(lds), "v"(gp)
                     : "memory");
    }
    asm volatile("s_wait_asynccnt 0x0" ::: "memory");
    __syncthreads();

    // Store phase: lanes run along channels -> coalesced NHWC writes.
#pragma unroll
    for (int j = 0; j < 4; ++j) {
        const int hw = hw0 + ty + j * 8;
        const int c  = c0 + tx;
        if (hw < HW) {
            d[(size_t)hw * 256 + c] = tile[tx][ty + j * 8];
        }
    }
}

// ---------------------------------------------------------------------------
// Stage 2: ROIAlign (aligned=True, OUT=14, SR=2) with FPN level assignment.
// One block per (box, output row). 64 threads, each owns 4 channels.
// NHWC=true : gathers are float4 loads from the transposed scratch (coalesced)
// NHWC=false: fallback, strided gathers directly from NCHW inputs
// ---------------------------------------------------------------------------
template <bool NHWC>
__device__ __forceinline__ float4 load4(const float* __restrict__ fb,
                                        int y, int x, int W, size_t HW, int t) {
    if (NHWC) {
        const float* p = fb + (((size_t)(y * W + x)) << 8) + 4 * t;  // * 256 ch
        return *reinterpret_cast<const float4*>(p);
    } else {
        const float* p = fb + (size_t)y * W + x;  // fb already at (n, 4t) channel
        float4 r;
        r.x = p[0];
        r.y = p[HW];
        r.z = p[2 * HW];
        r.w = p[3 * HW];
        return r;
    }
}

template <bool NHWC>
__global__ __launch_bounds__(64) void roi_kernel(const float* __restrict__ boxes,
                                                 const float* __restrict__ l0,
                                                 const float* __restrict__ l1,
                                                 const float* __restrict__ l2,
                                                 const float* __restrict__ l3,
                                                 float* __restrict__ out) {
    const int m  = blockIdx.x;   // box
    const int py = blockIdx.y;   // output row (y)
    const int t  = threadIdx.x;  // channel group: channels [4t, 4t+4)

    const float bi  = boxes[m * 5 + 0];
    const float bx0 = boxes[m * 5 + 1];
    const float by0 = boxes[m * 5 + 2];
    const float bx1 = boxes[m * 5 + 3];
    const float by1 = boxes[m * 5 + 4];
    const int   n   = (int)bi;

    // FPN level: floor(4 + log2(sqrt(area)/224 + eps)) clipped to [2,5], -2
    const float size = sqrtf((bx1 - bx0) * (by1 - by0));
    float lf = floorf(4.0f + log2f(size / 224.0f + 2.220446049250313e-16f));
    lf = fminf(fmaxf(lf, 2.0f), 5.0f);
    const int lvl = (int)lf - 2;

    int H, W;
    float sc;
    const float* f;
    switch (lvl) {
        case 0:  H = 200; W = 200; sc = 0.25f;    f = l0; break;
        case 1:  H = 100; W = 100; sc = 0.125f;   f = l1; break;
        case 2:  H = 50;  W = 50;  sc = 0.0625f;  f = l2; break;
        default: H = 25;  W = 25;  sc = 0.03125f; f = l3; break;
    }
    const size_t HW = (size_t)H * W;
    const float* fb = NHWC ? (f + (size_t)n * HW * 256)
                           : (f + ((size_t)n * 256 + 4 * (size_t)t) * HW);

    // aligned=True ROI geometry
    const float x0 = bx0 * sc - 0.5f;
    const float y0 = by0 * sc - 0.5f;
    const float bw = ((bx1 * sc - 0.5f) - x0) * (1.0f / 14.0f);
    const float bh = ((by1 * sc - 0.5f) - y0) * (1.0f / 14.0f);

    // Two y-samples for this output row (SR=2)
    bool  yv[2];
    int   ylo[2], yhi[2];
    float lyv[2];
#pragma unroll
    for (int iy = 0; iy < 2; ++iy) {
        const float y = y0 + ((float)py + ((float)iy + 0.5f) * 0.5f) * bh;
        yv[iy] = (y > -1.0f) && (y < (float)H);
        const float yc = fminf(fmaxf(y, 0.0f), (float)(H - 1));
        const float fl = floorf(yc);
        lyv[iy] = yc - fl;
        ylo[iy] = (int)fl;
        const int hi = ylo[iy] + 1;
        yhi[iy] = hi < (H - 1) ? hi : (H - 1);
    }

    float* op = out + ((size_t)m * 256 + 4 * (size_t)t) * 196 + (size_t)py * 14;

    for (int px = 0; px < 14; ++px) {
        // Two x-samples for this output column
        bool  xv[2];
        int   xlo[2], xhi[2];
        float lxv[2];
#pragma unroll
        for (int ix = 0; ix < 2; ++ix) {
            const float x = x0 + ((float)px + ((float)ix + 0.5f) * 0.5f) * bw;
            xv[ix] = (x > -1.0f) && (x < (float)W);
            const float xc = fminf(fmaxf(x, 0.0f), (float)(W - 1));
            const float fl = floorf(xc);
            lxv[ix] = xc - fl;
            xlo[ix] = (int)fl;
            const int hi = xlo[ix] + 1;
            xhi[ix] = hi < (W - 1) ? hi : (W - 1);
        }

        float4 acc = make_float4(0.f, 0.f, 0.f, 0.f);
#pragma unroll
        for (int iy = 0; iy < 2; ++iy) {
#pragma unroll
            for (int ix = 0; ix < 2; ++ix) {
                if (yv[iy] && xv[ix]) {
                    const float ly = lyv[iy];
                    const float lx = lxv[ix];
                    const float wll = (1.f - ly) * (1.f - lx);
                    const float wlh = (1.f - ly) * lx;
                    const float whl = ly * (1.f - lx);
                    const float whh = ly * lx;
                    const float4 vll = load4<NHWC>(fb, ylo[iy], xlo[ix], W, HW, t);
                    const float4 vlh = load4<NHWC>(fb, ylo[iy], xhi[ix], W, HW, t);
                    const float4 vhl = load4<NHWC>(fb, yhi[iy], xlo[ix], W, HW, t);
                    const float4 vhh = load4<NHWC>(fb, yhi[iy], xhi[ix], W, HW, t);
                    acc.x += ((wll * vll.x + wlh * vlh.x) + whl * vhl.x) + whh * vhh.x;
                    acc.y += ((wll * vll.y + wlh * vlh.y) + whl * vhl.y) + whh * vhh.y;
                    acc.z += ((wll * vll.z + wlh * vlh.z) + whl * vhl.z) + whh * vhh.z;
                    acc.w += ((wll * vll.w + wlh * vlh.w) + whl * vhl.w) + whh * vhh.w;
                }
            }
        }
        op[px + 0]   = acc.x * 0.25f;
        op[px + 196] = acc.y * 0.25f;
        op[px + 392] = acc.z * 0.25f;
        op[px + 588] = acc.w * 0.25f;
    }
}

// ---------------------------------------------------------------------------
// Host launcher
// Inputs: d_in[0..3] = f0..f3 (fp32 NCHW), d_in[4] = boxes [M,5] fp32
// Output: [M, 256, 14, 14] fp32
// ---------------------------------------------------------------------------
extern "C" void kernel_launch(void* const* d_in, const int* in_sizes, int n_in,
                              void* d_out, int out_size, void* d_ws, size_t ws_size,
                              hipStream_t stream) {
    const float* f0    = (const float*)d_in[0];
    const float* f1    = (const float*)d_in[1];
    const float* f2    = (const float*)d_in[2];
    const float* f3    = (const float*)d_in[3];
    const float* boxes = (const float*)d_in[4];
    float*       out   = (float*)d_out;
    const int    M     = in_sizes[4] / 5;

    // NHWC scratch layout (floats): lvl0 | lvl1 | lvl2 | lvl3
    const size_t F0 = (size_t)2 * 40000 * 256;  // 200x200
    const size_t F1 = (size_t)2 * 10000 * 256;  // 100x100
    const size_t F2 = (size_t)2 * 2500  * 256;  //  50x50
    const size_t F3 = (size_t)2 * 625   * 256;  //  25x25
    const size_t NEED = (F0 + F1 + F2 + F3) * sizeof(float);  // ~108.8 MB

    if (ws_size >= NEED) {
        float* t0 = (float*)d_ws;
        float* t1 = t0 + F0;
        float* t2 = t1 + F1;
        float* t3 = t2 + F2;

        const dim3 blk(32, 8, 1);
        nchw_to_nhwc<<<dim3((40000 + 31) / 32, 8, 2), blk, 0, stream>>>(f0, t0, 40000);
        nchw_to_nhwc<<<dim3((10000 + 31) / 32, 8, 2), blk, 0, stream>>>(f1, t1, 10000);
        nchw_to_nhwc<<<dim3((2500  + 31) / 32, 8, 2), blk, 0, stream>>>(f2, t2, 2500);
        nchw_to_nhwc<<<dim3((625   + 31) / 32, 8, 2), blk, 0, stream>>>(f3, t3, 625);

        roi_kernel<true><<<dim3(M, 14, 1), 64, 0, stream>>>(boxes, t0, t1, t2, t3, out);
    } else {
        // Scratch too small: gather straight from NCHW inputs.
        roi_kernel<false><<<dim3(M, 14, 1), 64, 0, stream>>>(boxes, f0, f1, f2, f3, out);
    }
}